// SiameseNetwork_49168785605520
// MI455X (gfx1250) — compile-verified
//
#include <hip/hip_runtime.h>
#include <math.h>

typedef __attribute__((ext_vector_type(2))) float v2f;
typedef __attribute__((ext_vector_type(8))) float v8f;

#define BPB 8            // batch elements per block
#define ROWS 16          // cat rows per block (BPB * 2 branches)
#define CAT_STRIDE 1028  // 1024 + 4-float pad -> conflict-free WMMA A-frag reads
#define R_STRIDE 305     // 304 + 1 pad
#define NNB 63           // neighbors
#define KDIM 1024        // cat width
#define NOUT 300
#define NTILES 19        // ceil(300/16)

__device__ __forceinline__ float wave_sum(float x) {
#pragma unroll
  for (int o = 16; o > 0; o >>= 1) x += __shfl_xor(x, o, 32);
  return x;
}
__device__ __forceinline__ float wave_max(float x) {
#pragma unroll
  for (int o = 16; o > 0; o >>= 1) x = fmaxf(x, __shfl_xor(x, o, 32));
  return x;
}

__global__ __launch_bounds__(256) void siamese_fused(
    const float* __restrict__ inp,   // (B, 2, 64, 512) f32
    const float* __restrict__ vvec,  // (63,) f32
    const float* __restrict__ Wm,    // (300, 1024) f32 row-major
    const float* __restrict__ bias,  // (300,) f32
    float* __restrict__ out)         // (B,) f32
{
  __shared__ float s_cat[ROWS * CAT_STRIDE];
  __shared__ float s_r[ROWS * R_STRIDE];
  __shared__ float s_p[8][64];

  const int tid  = threadIdx.x;
  const int lane = tid & 31;
  // wave id is uniform across the wave: pin it into an SGPR so loop control,
  // base addresses and LDS row offsets become scalar (and EXEC stays all-1s
  // around every WMMA, as the ISA requires).
  const int wave = __builtin_amdgcn_readfirstlane(tid >> 5);
  const int b0   = blockIdx.x * BPB;

  // ---------- Phase 1+2: attention + context; build cat rows in LDS ----------
  for (int t = 0; t < 2; ++t) {
    const int tt = wave * 2 + t;          // cat row index in block (uniform)
    const int b  = b0 + (tt >> 1);
    const int s  = tt & 1;
    const float4* base =
        (const float4*)(inp + (size_t)(b * 2 + s) * (64 * 512));

    // node: lane owns d = j*128 + lane*4 .. +3  (coalesced 512B/wave loads)
    float4 nd[4];
#pragma unroll
    for (int j = 0; j < 4; ++j) nd[j] = base[j * 32 + lane];

    // attention logits: att[k] = dot(neigh_k, node)
    float a0 = 0.f, a1 = 0.f;
    for (int k = 0; k < NNB; ++k) {
      const float4* nb = base + (k + 1) * 128;  // row k+1 (512 f32 = 128 f4)
      float acc = 0.f;
#pragma unroll
      for (int j = 0; j < 4; ++j) {
        float4 q = nb[j * 32 + lane];
        acc = fmaf(nd[j].x, q.x, acc);
        acc = fmaf(nd[j].y, q.y, acc);
        acc = fmaf(nd[j].z, q.z, acc);
        acc = fmaf(nd[j].w, q.w, acc);
      }
      float red = wave_sum(acc);
      if (k < 32) { if (lane == k)      a0 = red; }
      else        { if (lane == k - 32) a1 = red; }
    }

    // mask (att==0 -> -9999) + softmax across the 63 logits
    float a0m = (a0 == 0.f) ? -9999.f : a0;
    const bool hv = (lane < 31);               // lane31's second slot is k=63: invalid
    float a1m = hv ? ((a1 == 0.f) ? -9999.f : a1) : -3.4e38f;
    float mx  = wave_max(fmaxf(a0m, a1m));
    float e0  = __expf(a0m - mx);
    float e1  = hv ? __expf(a1m - mx) : 0.f;
    float Ssum = wave_sum(e0 + e1);
    float inv  = 1.f / Ssum;
    float v0 = vvec[lane];
    float v1 = hv ? vvec[32 + lane] : 0.f;
    s_p[wave][lane]      = e0 * inv * v0;      // fold v[k] into the weight
    s_p[wave][32 + lane] = e1 * inv * v1;

    // context_d = sum_k p[k] * neigh[k][d]   (re-stream rows; L2-hot)
    float4 cx[4] = {};
    for (int k = 0; k < NNB; ++k) {
      float pk = s_p[wave][k];                 // LDS broadcast
      const float4* nb = base + (k + 1) * 128;
#pragma unroll
      for (int j = 0; j < 4; ++j) {
        float4 q = nb[j * 32 + lane];
        cx[j].x = fmaf(pk, q.x, cx[j].x);
        cx[j].y = fmaf(pk, q.y, cx[j].y);
        cx[j].z = fmaf(pk, q.z, cx[j].z);
        cx[j].w = fmaf(pk, q.w, cx[j].w);
      }
    }

    // cat row = [node(512) | context(512)]
    float* cr = &s_cat[tt * CAT_STRIDE];
#pragma unroll
    for (int j = 0; j < 4; ++j) {
      *(float4*)&cr[j * 128 + lane * 4]       = nd[j];
      *(float4*)&cr[512 + j * 128 + lane * 4] = cx[j];
    }
  }
  __syncthreads();

  // ---------- Phase 3: r = cat @ W^T + bias via V_WMMA_F32_16X16X4_F32 ----------
  // A (16x4 f32): lanes 0-15 hold row M=lane, K=k0..k0+1; lanes 16-31 hold K=k0+2..k0+3
  // B (4x16 f32): lane n holds col n; VGPR0/1 = K pairs mirroring A halves
  // C/D (16x16 f32): VGPR i -> M = i + 8*(lane>>4), N = lane&15
  {
    const int lr = lane & 15;
    const int hs = lane >> 4;
    const int cb = 2 * hs;
    for (int nt = wave; nt < NTILES; nt += 8) {   // scalar loop (nt uniform)
      const int n0 = nt * 16;
      const int wr = n0 + lr;
      const float* wrow = Wm + (size_t)((wr < NOUT) ? wr : (NOUT - 1)) * KDIM;
      float bv = (wr < NOUT) ? bias[wr] : 0.f;
      v8f acc;
#pragma unroll
      for (int i = 0; i < 8; ++i) acc[i] = bv;

#pragma unroll 8
      for (int k0 = 0; k0 < KDIM; k0 += 4) {
        v2f a  = *(const v2f*)&s_cat[lr * CAT_STRIDE + k0 + cb];
        v2f bw = *(const v2f*)&wrow[k0 + cb];
        acc = __builtin_amdgcn_wmma_f32_16x16x4_f32(
            /*neg_a=*/false, a, /*neg_b=*/false, bw,
            /*c_mod=*/(short)0, acc, /*reuse_a=*/false, /*reuse_b=*/false);
      }
#pragma unroll
      for (int i = 0; i < 8; ++i)
        s_r[(i + 8 * hs) * R_STRIDE + n0 + lr] = acc[i];
    }
  }
  __syncthreads();

  // ---------- Phase 4: cosine similarity per batch element ----------
  {
    const float* r0 = &s_r[(wave * 2 + 0) * R_STRIDE];
    const float* r1 = &s_r[(wave * 2 + 1) * R_STRIDE];
    float dp = 0.f, q0 = 0.f, q1 = 0.f;
    for (int n = lane; n < NOUT; n += 32) {
      float x = r0[n], y = r1[n];
      dp = fmaf(x, y, dp);
      q0 = fmaf(x, x, q0);
      q1 = fmaf(y, y, q1);
    }
    dp = wave_sum(dp);
    q0 = wave_sum(q0);
    q1 = wave_sum(q1);
    if (lane == 0) {
      float n0n = fmaxf(sqrtf(q0), 1e-8f);
      float n1n = fmaxf(sqrtf(q1), 1e-8f);
      out[b0 + wave] = dp / (n0n * n1n);
    }
  }
}

extern "C" void kernel_launch(void* const* d_in, const int* in_sizes, int n_in,
                              void* d_out, int out_size, void* d_ws, size_t ws_size,
                              hipStream_t stream) {
  const float* inp  = (const float*)d_in[0];  // (8192,2,64,512)
  const float* vvec = (const float*)d_in[1];  // (63,)
  const float* Wm   = (const float*)d_in[2];  // (300,1024)
  const float* bias = (const float*)d_in[3];  // (300,)
  float* out = (float*)d_out;                 // (8192,)
  const int blocks = out_size / BPB;          // 1024
  siamese_fused<<<blocks, 256, 0, stream>>>(inp, vvec, Wm, bias, out);
}